// MemoryModule_8160437863135
// MI455X (gfx1250) — compile-verified
//
#include <hip/hip_runtime.h>
#include <hip/hip_bf16.h>
#include <math.h>

typedef __attribute__((ext_vector_type(2))) float v2f;
typedef __attribute__((ext_vector_type(8))) float v8f;

#define DIM   128
#define NUMS  2048
#define KTOP  129
#define DPAD  (DIM + 4)    // pad so A-fragment LDS reads spread across banks
#define SPAD  (NUMS + 4)   // pad so GEMM2 A-fragment reads spread across banks

// Fused: GEMM1 (scores) -> sigmoid -> exact top-129 mean -> GEMM2 (augment) + concat.
// One workgroup = 16 (b,t) rows. 256 threads = 8 wave32.
__global__ __launch_bounds__(256) void memmod_kernel(const float* __restrict__ data,
                                                     const float* __restrict__ weight,
                                                     float* __restrict__ out) {
    __shared__ float sData[16][DPAD];   // ~8.4 KB
    __shared__ float sS[16][SPAD];      // ~131.3 KB score tile

    const int wg   = blockIdx.x;        // 0..1023
    const int tid  = threadIdx.x;       // 0..255
    const int lane = tid & 31;
    const int wave = tid >> 5;          // 0..7
    const int rowBase = wg * 16;        // global (b*512+t) row of first tile row

    // ---- stage data tile in LDS (128-bit transfers) -----------------------
    {
        const float4* src = reinterpret_cast<const float4*>(data + (size_t)rowBase * DIM);
        for (int i = tid; i < 16 * DIM / 4; i += 256) {
            const int m  = i >> 5;          // DIM/4 = 32 float4 per row
            const int d4 = (i & 31) << 2;
            *reinterpret_cast<float4*>(&sData[m][d4]) = src[i];
        }
    }
    __syncthreads();

    // ---- A fragments for 16x128 data tile (f32 WMMA 16x16x4 layout) -------
    // lanes 0-15: M=lane,  K={4s+0,4s+1}; lanes 16-31: M=lane-16, K={4s+2,4s+3}
    const int mA   = lane & 15;
    const int kOff = (lane < 16) ? 0 : 2;
    v2f afrag[32];
#pragma unroll
    for (int s = 0; s < 32; ++s) {
        afrag[s] = *reinterpret_cast<const v2f*>(&sData[mA][4 * s + kOff]);
    }

    // ---- GEMM1: scores = sigmoid(data . weight^T / sqrt(128)) -------------
    // Two N-tiles per pass -> two independent WMMA accumulator chains.
    const float scale = 0.08838834764831845f; // 1/sqrt(128)
    const int  nLane  = lane & 15;
    const int  mBase  = (lane < 16) ? 0 : 8;
    for (int j = 0; j < 16; j += 2) {
        const int n0 = (wave * 16 + j) * 16;     // this wave's N-tile pair
        const int n1 = n0 + 16;
        const v2f* wrow0 = reinterpret_cast<const v2f*>(weight + (size_t)(n0 + nLane) * DIM + kOff);
        const v2f* wrow1 = reinterpret_cast<const v2f*>(weight + (size_t)(n1 + nLane) * DIM + kOff);
        v8f c0 = {}, c1 = {};
#pragma unroll
        for (int s = 0; s < 32; ++s) {
            v2f b0 = wrow0[2 * s];
            v2f b1 = wrow1[2 * s];
            c0 = __builtin_amdgcn_wmma_f32_16x16x4_f32(false, afrag[s], false, b0,
                                                       (short)0, c0, false, false);
            c1 = __builtin_amdgcn_wmma_f32_16x16x4_f32(false, afrag[s], false, b1,
                                                       (short)0, c1, false, false);
        }
        // C layout: VGPR r -> (M = r (+8 for hi lanes), N = lane&15)
#pragma unroll
        for (int r = 0; r < 8; ++r) {
            float x0 = c0[r] * scale;
            float x1 = c1[r] * scale;
            sS[mBase + r][n0 + nLane] = 1.0f / (1.0f + __expf(-x0));
            sS[mBase + r][n1 + nLane] = 1.0f / (1.0f + __expf(-x1));
        }
    }
    __syncthreads();

    // ---- exact top-129 mean per row (radix select on float bits) ----------
    // Each wave handles rows {wave, wave+8}; 64 score values live in VGPRs.
    float msc[2];
#pragma unroll
    for (int rr = 0; rr < 2; ++rr) {
        const int m = wave + 8 * rr;
        float v[64];
#pragma unroll
        for (int i = 0; i < 64; ++i) v[i] = sS[m][lane + 32 * i];

        unsigned thr = 0u;                 // scores in (0,1): positive floats order as uints
        for (int bit = 30; bit >= 0; --bit) {
            const unsigned cand = thr | (1u << bit);
            int cnt = 0;
#pragma unroll
            for (int i = 0; i < 64; ++i) cnt += (__float_as_uint(v[i]) >= cand) ? 1 : 0;
#pragma unroll
            for (int off = 16; off >= 1; off >>= 1) cnt += __shfl_xor(cnt, off, 32);
            if (cnt >= KTOP) thr = cand;   // thr converges to exact 129th-largest bits
        }
        const float tval = __uint_as_float(thr);
        int cntGT = 0; float sum = 0.0f;
#pragma unroll
        for (int i = 0; i < 64; ++i) {
            if (v[i] > tval) { cntGT += 1; sum += v[i]; }
        }
#pragma unroll
        for (int off = 16; off >= 1; off >>= 1) {
            cntGT += __shfl_xor(cntGT, off, 32);
            sum   += __shfl_xor(sum,   off, 32);
        }
        const float mean = (sum + (float)(KTOP - cntGT) * tval) * (1.0f / (float)KTOP);
        msc[rr] = 1.0f - mean;
    }
    if (lane == 0) {
        out[rowBase + wave]     = msc[0];
        out[rowBase + wave + 8] = msc[1];
    }

    // ---- GEMM2 + concat, only for b < 16 (wg < 512) -----------------------
    if (wg < 512) {
        float* rq = out + 32 * 512;        // read_query base: [8192][256]
        const int n0   = wave * 16;        // this wave's N-tile of the 128-wide augment
        const int nIdx = n0 + nLane;
        // Two interleaved K-chains for WMMA ILP; summed at the end.
        v8f c0 = {}, c1 = {};
        for (int k0 = 0; k0 < NUMS; k0 += 8) {
            v2f a0 = *reinterpret_cast<const v2f*>(&sS[mA][k0 + kOff]);
            v2f a1 = *reinterpret_cast<const v2f*>(&sS[mA][k0 + 4 + kOff]);
            v2f b0, b1;
            b0.x = weight[(size_t)(k0 + kOff + 0) * DIM + nIdx];
            b0.y = weight[(size_t)(k0 + kOff + 1) * DIM + nIdx];
            b1.x = weight[(size_t)(k0 + 4 + kOff + 0) * DIM + nIdx];
            b1.y = weight[(size_t)(k0 + 4 + kOff + 1) * DIM + nIdx];
            c0 = __builtin_amdgcn_wmma_f32_16x16x4_f32(false, a0, false, b0,
                                                       (short)0, c0, false, false);
            c1 = __builtin_amdgcn_wmma_f32_16x16x4_f32(false, a1, false, b1,
                                                       (short)0, c1, false, false);
        }
#pragma unroll
        for (int r = 0; r < 8; ++r) {
            const size_t row = (size_t)(rowBase + mBase + r);
            rq[row * 256 + 128 + nIdx] = c0[r] + c1[r];
        }
        // first half of read_query = data[:16] (128-bit transfers)
        for (int i = tid; i < 16 * DIM / 4; i += 256) {
            const int m  = i >> 5;
            const int d4 = (i & 31) << 2;
            *reinterpret_cast<float4*>(&rq[(size_t)(rowBase + m) * 256 + d4]) =
                *reinterpret_cast<const float4*>(&sData[m][d4]);
        }
    }
}

extern "C" void kernel_launch(void* const* d_in, const int* in_sizes, int n_in,
                              void* d_out, int out_size, void* d_ws, size_t ws_size,
                              hipStream_t stream) {
    const float* data   = (const float*)d_in[0];  // [32,512,128] f32
    const float* weight = (const float*)d_in[1];  // [2048,128]   f32
    float* out = (float*)d_out;                   // [16384] mem_score ++ [8192*256] read_query

    dim3 grid(1024);   // 16384 rows / 16 rows per WG
    dim3 block(256);   // 8 wave32
    memmod_kernel<<<grid, block, 0, stream>>>(data, weight, out);
}